// BinaryRelativePositionEmbedding_53077205844632
// MI455X (gfx1250) — compile-verified
//
#include <hip/hip_runtime.h>

// -------- problem constants (from reference) --------
#define SEQ    2048
#define DM     64
#define NBITS  12
#define MREL   4095            // 2*SEQ - 1
#define MPAD   4096            // padded rows for 16-wide tiling
#define BH     32              // BATCH * HEADS

typedef __attribute__((ext_vector_type(2))) float v2f;
typedef __attribute__((ext_vector_type(8))) float v8f;

// ---------------------------------------------------------------------------
// Kernel A: build relative-position table R[m][d] = sum_b bit(pos_m, b)*emb[b][d]
// pos_m = m - (SEQ-1); two's-complement bits via arithmetic shift.
// Row MREL (pad) is written as zeros so the GEMM can read it safely.
// ---------------------------------------------------------------------------
__global__ __launch_bounds__(256) void build_relpos_kernel(
    const float* __restrict__ emb, float* __restrict__ relpos) {
  int idx = blockIdx.x * 256 + threadIdx.x;   // 0 .. MPAD*DM-1
  int m = idx >> 6;                           // row
  int d = idx & (DM - 1);                     // column
  float acc = 0.0f;
  int pos = m - (SEQ - 1);
  float valid = (m < MREL) ? 1.0f : 0.0f;
#pragma unroll
  for (int b = 0; b < NBITS; ++b) {
    float bit = (float)((pos >> b) & 1);
    acc += bit * emb[b * DM + d];
  }
  relpos[idx] = acc * valid;
}

// ---------------------------------------------------------------------------
// Kernel B: out[bh][l][m] = sum_d q[bh][l][d] * R[m][d]
// One wave32 computes a 16(l) x 64(m) slab: 4 subtiles of 16x16, each via
// 16 chained V_WMMA_F32_16X16X4_F32 over K = 64.
//
// ISA operand layouts (CDNA5 ISA, 32-bit A 16x4 / C 16x16):
//   A step k0: lane(0-15) row=lane, VGPR0/1 = K {4k0+0,4k0+1}; lanes 16-31
//              carry K {4k0+2,4k0+3}  -> per-lane contiguous float2 from q.
//   B step k0: mirrored with N striped across lanes -> per-lane contiguous
//              float2 from row-major R (R holds B^T, i.e. B[k][n] = R[n][k]).
//   C/D: VGPR v -> row v (lanes 0-15) / row v+8 (lanes 16-31), col = lane&15.
// ---------------------------------------------------------------------------
__global__ __launch_bounds__(256) void qrel_wmma_kernel(
    const float* __restrict__ q, const float* __restrict__ relpos,
    float* __restrict__ out) {
  const int lane = threadIdx.x & 31;
  const int wave = threadIdx.x >> 5;

  // job decomposition: 32 bh * 128 l-tiles * 64 m-groups = 262144 wave-jobs
  int job = blockIdx.x * 8 + wave;
  const int mg = job & 63;          // m-group: 64 columns of m
  const int lt = (job >> 6) & 127;  // l-tile: 16 rows of l
  const int bh = job >> 13;         // batch*head slice

  const int row  = lane & 15;       // M row (A) / N col (B) for this lane
  const int half = lane >> 4;

  // ---- preload A operands for all 16 K-steps (reused across 4 subtiles) ----
  const float* aptr =
      q + (size_t)(bh * SEQ + lt * 16 + row) * DM + 2 * half;
  v2f a[16];
#pragma unroll
  for (int k0 = 0; k0 < 16; ++k0)
    a[k0] = *(const v2f*)(aptr + 4 * k0);

  const int m0 = mg * 64;

#pragma unroll
  for (int sub = 0; sub < 4; ++sub) {
    const int col = m0 + sub * 16 + row;   // this lane's m column
    const float* bptr = relpos + (size_t)col * DM + 2 * half;

    v8f c = {};
#pragma unroll
    for (int k0 = 0; k0 < 16; ++k0) {
      v2f b = *(const v2f*)(bptr + 4 * k0);
      // (neg_a, A, neg_b, B, c_mod, C, reuse_a, reuse_b)
      c = __builtin_amdgcn_wmma_f32_16x16x4_f32(
          false, a[k0], false, b, (short)0, c, false, false);
    }

    // ---- store 16x16 tile; guard the single padded m column (m == 4095) ----
    if (col < MREL) {
      float* obase =
          out + (size_t)(bh * SEQ + lt * 16) * (size_t)MREL + (size_t)col;
#pragma unroll
      for (int v = 0; v < 8; ++v) {
        int r = v + half * 8;
        // streaming 1.07 GB: non-temporal so L2 keeps q/R hot
        __builtin_nontemporal_store(c[v], obase + (size_t)r * MREL);
      }
    }
  }
}

// ---------------------------------------------------------------------------
extern "C" void kernel_launch(void* const* d_in, const int* in_sizes, int n_in,
                              void* d_out, int out_size, void* d_ws, size_t ws_size,
                              hipStream_t stream) {
  const float* q   = (const float*)d_in[0];  // [2,16,2048,64]
  // d_in[1] (k) is unused by the reference
  const float* emb = (const float*)d_in[2];  // [12,64]
  float* out = (float*)d_out;                // [2,16,2048,4095]
  float* relpos = (float*)d_ws;              // MPAD*DM f32 = 1 MB scratch

  // Kernel A: 4096*64 elements, one thread each
  build_relpos_kernel<<<(MPAD * DM) / 256, 256, 0, stream>>>(emb, relpos);

  // Kernel B: 262144 wave-jobs / 8 waves per block
  qrel_wmma_kernel<<<262144 / 8, 256, 0, stream>>>(q, relpos, out);
}